// CRFLayer_72103910966149
// MI455X (gfx1250) — compile-verified
//
#include <hip/hip_runtime.h>
#include <stdint.h>

// Viterbi decode: B=1024, S=512, T=128.
// Forward: max-plus recurrence, VALU-bound (~34e9 VALU ops vs ~14us of HBM).
// transitions (64KB) cached in LDS via CDNA5 async bulk copy; v ping-pong in LDS.
// Backpointers stored as u8 (64MB) -> stays resident in 192MB L2 for backtrace.

#define VIT_B 1024
#define VIT_S 512
#define VIT_T 128
#define VIT_NB 2   // batches per block

extern __shared__ char vit_smem[];

__global__ __launch_bounds__(VIT_T) void viterbi_fwd_kernel(
    const float* __restrict__ feats,
    const float* __restrict__ trans,
    const float* __restrict__ start_t,
    const float* __restrict__ stop_t,
    uint8_t* __restrict__ bp,
    float* __restrict__ vfin)
{
  float* sT = (float*)vit_smem;                          // [T*T] transitions
  float* vb = (float*)(vit_smem + VIT_T * VIT_T * 4);    // [2][T*NB], v[buf][i*NB+n]

  const int j  = threadIdx.x;        // tag index 0..127
  const int b0 = blockIdx.x * VIT_NB;

  // ---- CDNA5 async bulk copy: transitions (64 KB) global -> LDS ----
  // 128 threads x 16B x 32 iters = 65536 B. LDS dest offset comes from the
  // low 32 bits of the shared pointer (aperture upper bits discarded by HW).
  {
    const char* gp0 = (const char*)trans + j * 16;
    uint32_t l0 = (uint32_t)(uintptr_t)sT + (uint32_t)(j * 16);
#pragma unroll
    for (int it = 0; it < 32; ++it) {
      uint32_t loff = l0 + it * 2048;
      const char* gp = gp0 + it * 2048;
      asm volatile("global_load_async_to_lds_b128 %0, %1, off"
                   :: "v"(loff), "v"(gp) : "memory");
    }
    asm volatile("s_wait_asynccnt 0" ::: "memory");
  }

  const size_t bstr  = (size_t)VIT_S * VIT_T;   // per-batch feats stride
  const size_t base0 = (size_t)b0 * bstr + (size_t)j;
  const size_t base1 = base0 + bstr;

  // v0 = feats[:,0,:] + start_transitions  (interleaved layout v[i*2+n])
  float st = start_t[j];
  vb[j * 2 + 0] = __builtin_nontemporal_load(&feats[base0]) + st;
  vb[j * 2 + 1] = __builtin_nontemporal_load(&feats[base1]) + st;
  float stopj = stop_t[j];
  __syncthreads();

  // software-pipelined feats loads (nontemporal: keep bp array hot in L2)
  float fc0 = __builtin_nontemporal_load(&feats[base0 + VIT_T]);
  float fc1 = __builtin_nontemporal_load(&feats[base1 + VIT_T]);

  int p = 0;
  for (int s = 1; s < VIT_S; ++s) {
    int sn = (s + 1 < VIT_S) ? s + 1 : s;   // clamp: harmless re-load on last iter
    float fn0 = __builtin_nontemporal_load(&feats[base0 + (size_t)sn * VIT_T]);
    float fn1 = __builtin_nontemporal_load(&feats[base1 + (size_t)sn * VIT_T]);

    const float2* vc = (const float2*)(vb + p * (2 * VIT_T));

    // i = 0 initializes (preserves jnp.argmax first-max tie-breaking via '>')
    float2 v0 = vc[0];
    float t0 = sT[j];
    float best0 = v0.x + t0;
    float best1 = v0.y + t0;
    int i0 = 0, i1 = 0;

#pragma unroll 8
    for (int i = 1; i < VIT_T; ++i) {
      float t = sT[i * VIT_T + j];     // ds_load_b32, conflict-free across lanes
      float2 vv = vc[i];               // ds_load_b64 broadcast (both batches)
      float a0 = vv.x + t;
      float a1 = vv.y + t;
      bool c0 = a0 > best0;
      bool c1 = a1 > best1;
      best0 = c0 ? a0 : best0;  i0 = c0 ? i : i0;
      best1 = c1 ? a1 : best1;  i1 = c1 ? i : i1;
    }

    // backpointers, u8, coalesced 128B per batch-row
    size_t bo = (size_t)(s - 1) * (VIT_B * VIT_T) + (size_t)b0 * VIT_T + (size_t)j;
    bp[bo]          = (uint8_t)i0;
    bp[bo + VIT_T]  = (uint8_t)i1;

    float* vn = vb + (1 - p) * (2 * VIT_T);
    vn[j * 2 + 0] = best0 + fc0;
    vn[j * 2 + 1] = best1 + fc1;
    __syncthreads();                 // single barrier/step (double-buffered v)
    p ^= 1;
    fc0 = fn0;
    fc1 = fn1;
  }

  // final scores with stop_transitions folded in
  vfin[(size_t)b0 * VIT_T + j]       = vb[p * (2 * VIT_T) + j * 2 + 0] + stopj;
  vfin[(size_t)(b0 + 1) * VIT_T + j] = vb[p * (2 * VIT_T) + j * 2 + 1] + stopj;
}

__global__ __launch_bounds__(128) void viterbi_bt_kernel(
    const uint8_t* __restrict__ bp,
    const float* __restrict__ vfin,
    int* __restrict__ out)
{
  int b = blockIdx.x * blockDim.x + threadIdx.x;
  if (b >= VIT_B) return;

  // last_tag = argmax_j (v_final + stop)   (first-max tie-breaking via '>')
  const float* v = vfin + (size_t)b * VIT_T;
  float best = v[0];
  int tag = 0;
#pragma unroll 4
  for (int jj = 1; jj < VIT_T; ++jj) {
    float x = v[jj];
    bool c = x > best;
    best = c ? x : best;
    tag = c ? jj : tag;
  }

  out[(size_t)b * VIT_S + (VIT_S - 1)] = tag;

  // pointer chase through L2-resident backpointers
  for (int k = VIT_S - 2; k >= 0; --k) {
    tag = (int)bp[(size_t)k * (VIT_B * VIT_T) + (size_t)b * VIT_T + (size_t)tag];
    out[(size_t)b * VIT_S + k] = tag;
  }
}

extern "C" void kernel_launch(void* const* d_in, const int* in_sizes, int n_in,
                              void* d_out, int out_size, void* d_ws, size_t ws_size,
                              hipStream_t stream) {
  (void)in_sizes; (void)n_in; (void)out_size; (void)ws_size;

  const float* feats   = (const float*)d_in[0];
  const float* trans   = (const float*)d_in[1];
  const float* start_t = (const float*)d_in[2];
  const float* stop_t  = (const float*)d_in[3];

  // workspace layout: bp u8[(S-1)*B*T]  (66,977,792 B)  then vfin f32[B*T]
  uint8_t* bp   = (uint8_t*)d_ws;
  float*   vfin = (float*)((char*)d_ws + (size_t)(VIT_S - 1) * VIT_B * VIT_T);
  int*     out  = (int*)d_out;

  // dynamic LDS: 64KB transitions + 2KB double-buffered v  (> 64KB static limit)
  size_t lds = (size_t)VIT_T * VIT_T * 4 + 2 * VIT_NB * VIT_T * 4;

  viterbi_fwd_kernel<<<VIT_B / VIT_NB, VIT_T, lds, stream>>>(
      feats, trans, start_t, stop_t, bp, vfin);
  viterbi_bt_kernel<<<(VIT_B + 127) / 128, 128, 0, stream>>>(bp, vfin, out);
}